// UpdateNodeBlock_52003464020803
// MI455X (gfx1250) — compile-verified
//
#include <hip/hip_runtime.h>
#include <math.h>

// ---------------- problem constants (match reference) ----------------
#define CCH      32          // channels
#define NBASIS   8           // radial basis functions
#define NPATHS   11          // TP paths
#define RCUT     5.0f
#define NORM_INV 0.1f        // 1/NORM_FACTOR
#define NSLOT    13          // 1 + 3 + 9 packed way components
#define NWAVES   2           // waves per block in node kernel

typedef float v2f __attribute__((ext_vector_type(2)));
typedef float v8f __attribute__((ext_vector_type(8)));

__device__ __forceinline__ float silu_f(float x) { return x / (1.0f + __expf(-x)); }

// ---------------- zero the scatter accumulators ----------------
__global__ void zero_kernel(float* __restrict__ p, int n) {
  int i = blockIdx.x * blockDim.x + threadIdx.x;
  int stride = gridDim.x * blockDim.x;
  for (; i < n; i += stride) p[i] = 0.0f;
}

// ---------------- edge kernel: one wave per edge, one lane per channel ----
// acc layout: acc[node][slot][channel], slot 0 = way0, 1..3 = way1, 4..12 = way2
__global__ __launch_bounds__(256) void edge_kernel(
    const float* __restrict__ rij,
    const float* __restrict__ x0, const float* __restrict__ x1,
    const float* __restrict__ x2,
    const int* __restrict__ idx_i, const int* __restrict__ idx_j,
    const float* __restrict__ Wrbf, const float* __restrict__ brbf,
    float* __restrict__ acc, int E)
{
  const int e = blockIdx.x * 8 + (threadIdx.x >> 5);
  const int c = threadIdx.x & 31;
  if (e >= E) return;

  const float rx = rij[(size_t)e*3+0], ry = rij[(size_t)e*3+1], rz = rij[(size_t)e*3+2];
  float d = sqrtf(rx*rx + ry*ry + rz*rz);
  d = fmaxf(d, 1e-6f);
  const float inv = 1.0f / d;
  const float rh[3] = { rx*inv, ry*inv, rz*inv };

  // Gaussian radial basis, centers linspace(0, RC, NB), width^2 = 0.25
  float rbf[NBASIS];
  #pragma unroll
  for (int b = 0; b < NBASIS; ++b) {
    const float dd = d - (RCUT / (float)(NBASIS - 1)) * (float)b;
    rbf[b] = __expf(-4.0f * dd * dd);
  }
  const float fc = 0.5f * (__cosf(3.14159265358979f * fminf(d, RCUT) / RCUT) + 1.0f);

  // radial filter MLP: filt[p][c] = (rbf @ W_rbf + b_rbf) * fc   (W_rbf fully L1-resident)
  float filt[NPATHS];
  #pragma unroll
  for (int p = 0; p < NPATHS; ++p) {
    float accf = brbf[p*CCH + c];
    #pragma unroll
    for (int b = 0; b < NBASIS; ++b)
      accf = fmaf(rbf[b], Wrbf[b*(NPATHS*CCH) + p*CCH + c], accf);
    filt[p] = accf * fc;
  }

  const int j = idx_j[e];
  const int i = idx_i[e];

  const float xj0 = x0[(size_t)j*CCH + c];
  float xj1[3];
  #pragma unroll
  for (int a = 0; a < 3; ++a) xj1[a] = x1[((size_t)j*CCH + c)*3 + a];
  float xj2[9];
  #pragma unroll
  for (int k = 0; k < 9; ++k) xj2[k] = x2[((size_t)j*CCH + c)*9 + k];

  const float x1r = xj1[0]*rh[0] + xj1[1]*rh[1] + xj1[2]*rh[2];
  float x2r[3];
  #pragma unroll
  for (int a = 0; a < 3; ++a)
    x2r[a] = xj2[a*3+0]*rh[0] + xj2[a*3+1]*rh[1] + xj2[a*3+2]*rh[2];
  const float rx2r = rh[0]*x2r[0] + rh[1]*x2r[1] + rh[2]*x2r[2];

  float m[NSLOT];
  m[0] = filt[0]*xj0 + filt[5]*x1r + filt[10]*rx2r;
  #pragma unroll
  for (int a = 0; a < 3; ++a)
    m[1+a] = filt[1]*xj0*rh[a] + filt[3]*xj1[a] + filt[6]*x1r*rh[a] + filt[8]*x2r[a];
  #pragma unroll
  for (int a = 0; a < 3; ++a)
    #pragma unroll
    for (int b = 0; b < 3; ++b)
      m[4+a*3+b] = filt[2]*xj0*rh[a]*rh[b] + filt[4]*xj1[a]*rh[b]
                 + filt[7]*xj2[a*3+b]      + filt[9]*x2r[a]*rh[b];

  float* base = acc + (size_t)i*NSLOT*CCH + c;
  #pragma unroll
  for (int s = 0; s < NSLOT; ++s)
    __hip_atomic_fetch_add(base + s*CCH, m[s]*NORM_INV,
                           __ATOMIC_RELAXED, __HIP_MEMORY_SCOPE_AGENT);
}

// ---------------- WMMA helpers (V_WMMA_F32_16X16X4_F32, chained K=32) -----
// B fragment (4x16): VGPR v holds K-row (4k+v) for lanes 0-15, K-row (4k+v+2) for 16-31.
__device__ __forceinline__ void load_wfrag(const float* __restrict__ W,
                                           int laneHi, int laneLo, v2f bf[2][8]) {
  #pragma unroll
  for (int t = 0; t < 2; ++t)
    #pragma unroll
    for (int kk = 0; kk < 8; ++kk) {
      const int kr  = 4*kk + 2*laneHi;
      const int col = 16*t + laneLo;
      bf[t][kk].x = W[kr*CCH + col];
      bf[t][kk].y = W[(kr+1)*CCH + col];
    }
}

// A fragment (16x4): lane holds row M=laneLo, elems (4k+2*laneHi, +1) -> one b64 load.
__device__ __forceinline__ void load_afrag(const float* __restrict__ acc,
                                           int n0, int s, int laneHi, int laneLo,
                                           v2f af[8]) {
  const float* row = acc + ((size_t)(n0 + laneLo)*NSLOT + s)*CCH + 2*laneHi;
  #pragma unroll
  for (int kk = 0; kk < 8; ++kk)
    af[kk] = *(const v2f*)(row + 4*kk);
}

__device__ __forceinline__ v8f mm16x16(const v2f af[8], const v2f bf[8]) {
  v8f cacc = {0.f,0.f,0.f,0.f,0.f,0.f,0.f,0.f};
  #pragma unroll
  for (int kk = 0; kk < 8; ++kk)
    cacc = __builtin_amdgcn_wmma_f32_16x16x4_f32(false, af[kk], false, bf[kk],
                                                 (short)0, cacc, false, false);
  return cacc;
}

// ---------------- node kernel: one wave per 16 nodes ----------------
__global__ __launch_bounds__(32*NWAVES) void node_kernel(
    const float* __restrict__ acc,
    const float* __restrict__ x0, const float* __restrict__ x1,
    const float* __restrict__ x2,
    const float* __restrict__ Wmix0, const float* __restrict__ Wmix1,
    const float* __restrict__ Wmix2, const float* __restrict__ coupling,
    const float* __restrict__ Wg0, const float* __restrict__ Wg1,
    const float* __restrict__ Wg2,
    const float* __restrict__ bg0, const float* __restrict__ bg1,
    const float* __restrict__ bg2,
    float* __restrict__ out, int Nn)
{
  __shared__ float y0s [NWAVES][16][33];       // y0 staged for gate-matmul A fragments
  __shared__ float ymat[NWAVES][12][16][33];   // Wmix matmul results, ways 1 & 2

  const int w      = threadIdx.x >> 5;
  const int lane   = threadIdx.x & 31;
  const int laneLo = lane & 15;
  const int laneHi = lane >> 4;
  const int n0     = (blockIdx.x * NWAVES + w) * 16;
  if (n0 >= Nn) return;   // whole wave exits together; active waves keep EXEC all-1s

  // coupling rows for this lane's two channels (tiles t=0,1)
  float cpv[8][2];
  #pragma unroll
  for (int k = 0; k < 8; ++k) {
    cpv[k][0] = coupling[k*CCH + laneLo];
    cpv[k][1] = coupling[k*CCH + 16 + laneLo];
  }

  v2f af[8];
  v2f bf[2][8];

  // ---- stage 1: y0 = a0 @ Wmix0 + elementwise couplings ----
  load_afrag(acc, n0, 0, laneHi, laneLo, af);
  load_wfrag(Wmix0, laneHi, laneLo, bf);
  v8f ylo = mm16x16(af, bf[0]);
  v8f yhi = mm16x16(af, bf[1]);

  #pragma unroll
  for (int t = 0; t < 2; ++t)
    #pragma unroll
    for (int r = 0; r < 8; ++r) {
      const int n = n0 + r + 8*laneHi;
      const int c = 16*t + laneLo;
      const float* ap = acc + (size_t)n*NSLOT*CCH + c;
      const float a0 = ap[0];
      float s1 = 0.f, s2 = 0.f;
      #pragma unroll
      for (int a = 0; a < 3; ++a) { const float v = ap[(1+a)*CCH]; s1 = fmaf(v, v, s1); }
      #pragma unroll
      for (int k = 0; k < 9; ++k) { const float v = ap[(4+k)*CCH]; s2 = fmaf(v, v, s2); }
      const float y0 = (t ? yhi[r] : ylo[r])
                     + cpv[0][t]*a0*a0 + cpv[3][t]*s1 + cpv[6][t]*s2;
      y0s[w][r + 8*laneHi][c] = y0;
    }
  __threadfence_block();

  // ---- stage 2: gates = silu(y0 @ Wg* + bg*); way-0 output ----
  v2f afy[8];
  #pragma unroll
  for (int kk = 0; kk < 8; ++kk) {
    const int k = 4*kk + 2*laneHi;
    afy[kk].x = y0s[w][laneLo][k];
    afy[kk].y = y0s[w][laneLo][k+1];
  }

  load_wfrag(Wg0, laneHi, laneLo, bf);
  v8f g0lo = mm16x16(afy, bf[0]);
  v8f g0hi = mm16x16(afy, bf[1]);
  #pragma unroll
  for (int t = 0; t < 2; ++t) {
    const float bias = bg0[16*t + laneLo];
    #pragma unroll
    for (int r = 0; r < 8; ++r) {
      const int n = n0 + r + 8*laneHi;
      const int c = 16*t + laneLo;
      const float z0 = silu_f((t ? g0hi[r] : g0lo[r]) + bias);
      out[((size_t)n*CCH + c)*NSLOT + 0] = x0[(size_t)n*CCH + c] + z0;
    }
  }

  v8f sg1lo, sg1hi, sg2lo, sg2hi;
  load_wfrag(Wg1, laneHi, laneLo, bf);
  {
    v8f glo = mm16x16(afy, bf[0]);
    v8f ghi = mm16x16(afy, bf[1]);
    const float b0 = bg1[laneLo], b1 = bg1[16 + laneLo];
    #pragma unroll
    for (int r = 0; r < 8; ++r) { sg1lo[r] = silu_f(glo[r] + b0); sg1hi[r] = silu_f(ghi[r] + b1); }
  }
  load_wfrag(Wg2, laneHi, laneLo, bf);
  {
    v8f glo = mm16x16(afy, bf[0]);
    v8f ghi = mm16x16(afy, bf[1]);
    const float b0 = bg2[laneLo], b1 = bg2[16 + laneLo];
    #pragma unroll
    for (int r = 0; r < 8; ++r) { sg2lo[r] = silu_f(glo[r] + b0); sg2hi[r] = silu_f(ghi[r] + b1); }
  }

  // ---- stage 3a: Wmix matmuls for all way-1/way-2 components -> LDS ----
  load_wfrag(Wmix1, laneHi, laneLo, bf);
  #pragma unroll
  for (int s = 1; s <= 12; ++s) {
    if (s == 4) load_wfrag(Wmix2, laneHi, laneLo, bf);
    load_afrag(acc, n0, s, laneHi, laneLo, af);
    v8f mlo = mm16x16(af, bf[0]);
    v8f mhi = mm16x16(af, bf[1]);
    #pragma unroll
    for (int t = 0; t < 2; ++t)
      #pragma unroll
      for (int r = 0; r < 8; ++r)
        ymat[w][s-1][r + 8*laneHi][16*t + laneLo] = (t ? mhi[r] : mlo[r]);
  }
  __threadfence_block();

  // ---- stage 3b: elementwise couplings + gating + residual, ways 1 & 2 ----
  #pragma unroll
  for (int t = 0; t < 2; ++t)
    #pragma unroll
    for (int r = 0; r < 8; ++r) {
      const int n  = n0 + r + 8*laneHi;
      const int c  = 16*t + laneLo;
      const int nr = r + 8*laneHi;
      const float* ap = acc + (size_t)n*NSLOT*CCH + c;
      const float a0 = ap[0];
      float a1v[3];
      #pragma unroll
      for (int a = 0; a < 3; ++a) a1v[a] = ap[(1+a)*CCH];
      float a2v[9];
      #pragma unroll
      for (int k = 0; k < 9; ++k) a2v[k] = ap[(4+k)*CCH];

      const float g1 = t ? sg1hi[r] : sg1lo[r];
      const float g2 = t ? sg2hi[r] : sg2lo[r];
      const size_t ob = ((size_t)n*CCH + c)*NSLOT;

      // way 1: y1 = mat + cp1*a0*a1 + cp5*(a2 . a1)
      #pragma unroll
      for (int a = 0; a < 3; ++a) {
        const float dot = a2v[a*3+0]*a1v[0] + a2v[a*3+1]*a1v[1] + a2v[a*3+2]*a1v[2];
        const float y = ymat[w][a][nr][c] + cpv[1][t]*a0*a1v[a] + cpv[5][t]*dot;
        out[ob + 1 + a] = x1[((size_t)n*CCH + c)*3 + a] + y*g1;
      }
      // way 2: y2 = mat + cp2*a0*a2 + cp4*a1 (x) a1 + cp7*(a2 @ a2)
      #pragma unroll
      for (int a = 0; a < 3; ++a)
        #pragma unroll
        for (int b = 0; b < 3; ++b) {
          const int k = a*3 + b;
          const float mm2 = a2v[a*3+0]*a2v[0*3+b] + a2v[a*3+1]*a2v[1*3+b]
                          + a2v[a*3+2]*a2v[2*3+b];
          const float y = ymat[w][3+k][nr][c] + cpv[2][t]*a0*a2v[k]
                        + cpv[4][t]*a1v[a]*a1v[b] + cpv[7][t]*mm2;
          out[ob + 4 + k] = x2[((size_t)n*CCH + c)*9 + k] + y*g2;
        }
    }
}

// ---------------- host launcher ----------------
extern "C" void kernel_launch(void* const* d_in, const int* in_sizes, int n_in,
                              void* d_out, int out_size, void* d_ws, size_t ws_size,
                              hipStream_t stream) {
  const float* rij   = (const float*)d_in[0];
  const float* x0    = (const float*)d_in[1];
  const float* x1    = (const float*)d_in[2];
  const float* x2    = (const float*)d_in[3];
  const int*   idx_i = (const int*)d_in[4];
  const int*   idx_j = (const int*)d_in[5];
  const float* Wrbf  = (const float*)d_in[6];
  const float* brbf  = (const float*)d_in[7];
  const float* Wmix0 = (const float*)d_in[8];
  const float* Wmix1 = (const float*)d_in[9];
  const float* Wmix2 = (const float*)d_in[10];
  const float* coup  = (const float*)d_in[11];
  const float* Wg0   = (const float*)d_in[12];
  const float* Wg1   = (const float*)d_in[13];
  const float* Wg2   = (const float*)d_in[14];
  const float* bg0   = (const float*)d_in[15];
  const float* bg1   = (const float*)d_in[16];
  const float* bg2   = (const float*)d_in[17];

  const int E = in_sizes[0] / 3;
  const int N = in_sizes[1] / CCH;

  float* acc = (float*)d_ws;                // N*13*32 f32 = ~33.3 MB scratch
  const int accN = N * NSLOT * CCH;

  zero_kernel<<<1024, 256, 0, stream>>>(acc, accN);

  const int eblocks = (E + 7) / 8;          // 8 edges (waves) per 256-thread block
  edge_kernel<<<eblocks, 256, 0, stream>>>(rij, x0, x1, x2, idx_i, idx_j,
                                           Wrbf, brbf, acc, E);

  const int ngroups = (N + 15) / 16;        // 16 nodes per wave (exact for N=20000)
  const int nblocks = (ngroups + NWAVES - 1) / NWAVES;
  node_kernel<<<nblocks, 32*NWAVES, 0, stream>>>(acc, x0, x1, x2,
                                                 Wmix0, Wmix1, Wmix2, coup,
                                                 Wg0, Wg1, Wg2, bg0, bg1, bg2,
                                                 (float*)d_out, N);
}